// graphconv_regression_confounded_31782757990680
// MI455X (gfx1250) — compile-verified
//
#include <hip/hip_runtime.h>

#define N_NODES  100000
#define N_EDGES  1600000
#define N_GRAPHS 64
#define DIM      128
#define IN_CH    4

typedef float v2f __attribute__((ext_vector_type(2)));
typedef float v8f __attribute__((ext_vector_type(8)));

// ---------------------------------------------------------------------------
// Edge aggregation: agg[dst] += h[src]  (scatter-add over edges)
// ---------------------------------------------------------------------------
__global__ __launch_bounds__(256) void edge_agg_c4(
    const float* __restrict__ h, const int* __restrict__ src,
    const int* __restrict__ dst, float* __restrict__ agg) {
  int e = blockIdx.x * blockDim.x + threadIdx.x;
  if (e >= N_EDGES) return;
  int s = src[e], d = dst[e];
  const float4 v = *(const float4*)(h + (size_t)s * IN_CH);
  float* a = agg + (size_t)d * IN_CH;
  atomicAdd(a + 0, v.x);
  atomicAdd(a + 1, v.y);
  atomicAdd(a + 2, v.z);
  atomicAdd(a + 3, v.w);
}

__global__ __launch_bounds__(256) void edge_agg_c128(
    const float* __restrict__ h, const int* __restrict__ src,
    const int* __restrict__ dst, float* __restrict__ agg) {
  // one thread per (edge, 4-channel chunk): 1.6M * 32 threads
  unsigned gid = blockIdx.x * blockDim.x + threadIdx.x;
  unsigned e  = gid >> 5;         // / 32 chunks
  unsigned ch = (gid & 31) << 2;  // chunk * 4
  if (e >= N_EDGES) return;
  int s = src[e], d = dst[e];
  const float4 v = *(const float4*)(h + (size_t)s * DIM + ch);
  float* a = agg + (size_t)d * DIM + ch;
  atomicAdd(a + 0, v.x);
  atomicAdd(a + 1, v.y);
  atomicAdd(a + 2, v.z);
  atomicAdd(a + 3, v.w);
}

// ---------------------------------------------------------------------------
// Fused dual GEMM: out = relu(A1 @ B1 + A2 @ B2 + bias)
//   A1,A2: [M x K]  B1,B2: [K x 128]  out: [M x 128]
// M = 100000 = 6250 * 16 exactly. One 16x16 tile per wave; block = 8 waves
// covering one 16-row strip across all N=128. fp32 WMMA (16x16x4), exact.
// ---------------------------------------------------------------------------
__global__ __launch_bounds__(256) void gemm_dual_wmma(
    const float* __restrict__ A1, const float* __restrict__ A2,
    const float* __restrict__ B1, const float* __restrict__ B2,
    const float* __restrict__ bias, float* __restrict__ out, int K) {
  const int lane  = threadIdx.x & 31;
  const int wv    = threadIdx.x >> 5;           // N-tile 0..7
  const int row0  = blockIdx.x << 4;            // M-tile base
  const int n     = lane & 15;
  const int col   = (wv << 4) + n;              // output column 0..127
  const int m     = lane & 15;                  // A row within tile
  const int khalf = (lane >> 4) << 1;           // 0 (lanes 0-15) / 2 (lanes 16-31)

  v8f acc = {};
  const float* a1row = A1 + (size_t)(row0 + m) * K + khalf;
  const float* a2row = A2 + (size_t)(row0 + m) * K + khalf;

  for (int k0 = 0; k0 < K; k0 += 4) {
    v2f a, b;
    a.x = a1row[k0];
    a.y = a1row[k0 + 1];
    b.x = B1[(size_t)(k0 + khalf) * DIM + col];
    b.y = B1[(size_t)(k0 + khalf + 1) * DIM + col];
    acc = __builtin_amdgcn_wmma_f32_16x16x4_f32(
        /*neg_a=*/false, a, /*neg_b=*/false, b,
        /*c_mod=*/(short)0, acc, /*reuse_a=*/false, /*reuse_b=*/false);
  }
  for (int k0 = 0; k0 < K; k0 += 4) {
    v2f a, b;
    a.x = a2row[k0];
    a.y = a2row[k0 + 1];
    b.x = B2[(size_t)(k0 + khalf) * DIM + col];
    b.y = B2[(size_t)(k0 + khalf + 1) * DIM + col];
    acc = __builtin_amdgcn_wmma_f32_16x16x4_f32(
        false, a, false, b, (short)0, acc, false, false);
  }

  const float bv   = bias[col];
  const int  rbase = row0 + ((lane >> 4) << 3);  // +8 for upper half-lanes
#pragma unroll
  for (int j = 0; j < 8; ++j) {
    float v = acc[j] + bv;
    out[(size_t)(rbase + j) * DIM + col] = v > 0.f ? v : 0.f;
  }
}

// ---------------------------------------------------------------------------
// Per-graph pooling: sum, max (values >= 0 post-ReLU -> int-bitcast max ok)
// ---------------------------------------------------------------------------
__global__ __launch_bounds__(128) void pool_kernel(
    const float* __restrict__ h, const int* __restrict__ batch,
    float* __restrict__ x_sum, int* __restrict__ x_max,
    int* __restrict__ counts) {
  int node = blockIdx.x;
  int c    = threadIdx.x;
  int g    = batch[node];
  float v  = h[(size_t)node * DIM + c];
  atomicAdd(&x_sum[g * DIM + c], v);
  atomicMax(&x_max[g * DIM + c], __float_as_int(v));  // v >= 0
  if (c == 0) atomicAdd(&counts[g], 1);
}

// ---------------------------------------------------------------------------
// Head: concat([x_max, x_mean, relu(metadata*convm_w+convm_b)]) -> fc -> fc2
// ---------------------------------------------------------------------------
__global__ __launch_bounds__(128) void head_kernel(
    const float* __restrict__ x_sum, const int* __restrict__ x_max,
    const int* __restrict__ counts, const float* __restrict__ metadata,
    const float* __restrict__ convm_w, const float* __restrict__ convm_b,
    const float* __restrict__ fc_w, const float* __restrict__ fc_b,
    const float* __restrict__ fc2_w, const float* __restrict__ fc2_b,
    float* __restrict__ out) {
  const int g = blockIdx.x;
  const int t = threadIdx.x;  // 0..127
  __shared__ float xc[2 * DIM + 4];
  __shared__ float red[128];

  float cnt = (float)counts[g];
  float inv = 1.0f / fmaxf(cnt, 1.0f);
  xc[t]       = (cnt > 0.f) ? __int_as_float(x_max[g * DIM + t]) : 0.f;
  xc[DIM + t] = x_sum[g * DIM + t] * inv;
  if (t < 4) {
    float mv = metadata[g] * convm_w[t] + convm_b[t];
    xc[2 * DIM + t] = mv > 0.f ? mv : 0.f;
  }
  __syncthreads();

  float s = fc_b[t];
  for (int i = 0; i < 2 * DIM + 4; ++i) s += xc[i] * fc_w[i * DIM + t];
  s = s > 0.f ? s : 0.f;

  red[t] = s * fc2_w[t];
  __syncthreads();
  for (int off = 64; off > 0; off >>= 1) {
    if (t < off) red[t] += red[t + off];
    __syncthreads();
  }
  if (t == 0) out[g] = red[0] + fc2_b[0];
}

// ---------------------------------------------------------------------------
// Launch
// ---------------------------------------------------------------------------
extern "C" void kernel_launch(void* const* d_in, const int* in_sizes, int n_in,
                              void* d_out, int out_size, void* d_ws, size_t ws_size,
                              hipStream_t stream) {
  const float* x        = (const float*)d_in[0];
  const float* metadata = (const float*)d_in[1];
  const int*   eidx     = (const int*)d_in[2];
  const int*   batch    = (const int*)d_in[3];
  const float* w_rel[4]  = {(const float*)d_in[4],  (const float*)d_in[7],
                            (const float*)d_in[10], (const float*)d_in[13]};
  const float* b_rel[4]  = {(const float*)d_in[5],  (const float*)d_in[8],
                            (const float*)d_in[11], (const float*)d_in[14]};
  const float* w_root[4] = {(const float*)d_in[6],  (const float*)d_in[9],
                            (const float*)d_in[12], (const float*)d_in[15]};
  const float* convm_w = (const float*)d_in[16];
  const float* convm_b = (const float*)d_in[17];
  const float* fc_w    = (const float*)d_in[18];
  const float* fc_b    = (const float*)d_in[19];
  const float* fc2_w   = (const float*)d_in[20];
  const float* fc2_b   = (const float*)d_in[21];
  float* out = (float*)d_out;

  const int* src = eidx;
  const int* dst = eidx + N_EDGES;

  // workspace carve-up
  const size_t FEAT_BYTES = (size_t)N_NODES * DIM * sizeof(float);  // 51.2 MB
  char* ws   = (char*)d_ws;
  float* agg = (float*)(ws);
  float* hA  = (float*)(ws + FEAT_BYTES);
  float* hB  = (float*)(ws + 2 * FEAT_BYTES);
  char*  sm  = ws + 3 * FEAT_BYTES;
  float* x_sum  = (float*)(sm);
  int*   x_max  = (int*)(sm + N_GRAPHS * DIM * sizeof(float));
  int*   counts = (int*)(sm + 2 * N_GRAPHS * DIM * sizeof(float));

  const int gemm_grid = N_NODES / 16;  // 6250, exact

  // ---- layer 1 (din = 4) ----
  hipMemsetAsync(agg, 0, (size_t)N_NODES * IN_CH * sizeof(float), stream);
  edge_agg_c4<<<(N_EDGES + 255) / 256, 256, 0, stream>>>(x, src, dst, agg);
  gemm_dual_wmma<<<gemm_grid, 256, 0, stream>>>(agg, x, w_rel[0], w_root[0],
                                                b_rel[0], hA, IN_CH);
  // ---- layers 2..4 (din = 128) ----
  float* hin  = hA;
  float* hout = hB;
  const unsigned agg_grid128 = (unsigned)(((size_t)N_EDGES * 32 + 255) / 256);
  for (int l = 1; l < 4; ++l) {
    hipMemsetAsync(agg, 0, FEAT_BYTES, stream);
    edge_agg_c128<<<agg_grid128, 256, 0, stream>>>(hin, src, dst, agg);
    gemm_dual_wmma<<<gemm_grid, 256, 0, stream>>>(agg, hin, w_rel[l], w_root[l],
                                                  b_rel[l], hout, DIM);
    float* tmp = hin; hin = hout; hout = tmp;
  }
  // final activations are in `hin`

  // ---- pooling ----
  hipMemsetAsync(sm, 0, 2 * N_GRAPHS * DIM * sizeof(float) + N_GRAPHS * sizeof(int),
                 stream);
  pool_kernel<<<N_NODES, 128, 0, stream>>>(hin, batch, x_sum, x_max, counts);

  // ---- head ----
  head_kernel<<<N_GRAPHS, 128, 0, stream>>>(x_sum, x_max, counts, metadata,
                                            convm_w, convm_b, fc_w, fc_b,
                                            fc2_w, fc2_b, out);
}